// LDA_61873298866375
// MI455X (gfx1250) — compile-verified
//
#include <hip/hip_runtime.h>
#include <hip/hip_bf16.h>
#include <math.h>

#define B_SZ   256
#define Mdim   272
#define Ndim   1089
#define IMG    33
#define CH     32
#define NPIX   (B_SZ * Ndim)          // 278784 (divisible by 64)
#define NT     ((Ndim + 15) / 16)     // 69 column tiles
#define KCONV  (CH * 9)               // 288
#define LDX    1104                   // padded row stride = 69*16 (full tiles fit)
#define LDPHI  1104                   // padded Phi cols
#define LDPTP  1120                   // padded PhiTPhi cols
#define RPTP   1104                   // padded PhiTPhi rows (full tiles fit)
#define PPL    1225                   // 35*35 padded plane
#define DELTA_C 0.01f
#define SIGMA_C 15000.0f

typedef __attribute__((ext_vector_type(2))) float v2f;
typedef __attribute__((ext_vector_type(8))) float v8f;

// ---------------------------------------------------------------- WMMA core
__device__ __forceinline__ v8f wmma4(v2f a, v2f b, v8f c) {
  return __builtin_amdgcn_wmma_f32_16x16x4_f32(false, a, false, b, (short)0, c,
                                               false, false);
}

// ---------------------------------------------------------------- activation
__device__ __forceinline__ float actf(float v) {
  float smooth = v * v * (1.0f / (4.0f * DELTA_C)) + 0.5f * v + DELTA_C * 0.25f;
  float r = fmaxf(v, 0.0f);
  return (fabsf(v) > DELTA_C) ? r : smooth;
}
__device__ __forceinline__ float actderf(float v) {
  float s = v * (1.0f / (2.0f * DELTA_C)) + 0.5f;
  float h = (v > 0.0f) ? 1.0f : 0.0f;
  return (fabsf(v) > DELTA_C) ? h : s;
}

// ================================================================ pad/init
__global__ void k_pad_phi(const float* __restrict__ Phi, float* __restrict__ Phip) {
  int i = blockIdx.x * blockDim.x + threadIdx.x;          // 272*1104
  if (i >= Mdim * LDPHI) return;
  int row = i / LDPHI, col = i % LDPHI;
  Phip[i] = (col < Ndim) ? Phi[row * Ndim + col] : 0.0f;
}
__global__ void k_pad_qinit(const float* __restrict__ Q, float* __restrict__ Qp) {
  int i = blockIdx.x * blockDim.x + threadIdx.x;          // 1104*272
  if (i >= LDPHI * Mdim) return;
  int row = i / Mdim, col = i % Mdim;
  Qp[i] = (row < Ndim) ? Q[row * Mdim + col] : 0.0f;
}
// zero borders of the two slabs used as conv inputs (G, H)
__global__ void k_zero_borders(float* __restrict__ G, float* __restrict__ H) {
  int i = blockIdx.x * blockDim.x + threadIdx.x;          // 8192*136
  const int NB = 136;
  if (i >= B_SZ * CH * NB) return;
  int plane = i / NB, e = i % NB;
  int y, x;
  if (e < 35)      { y = 0;  x = e; }
  else if (e < 70) { y = 34; x = e - 35; }
  else             { int r = e - 70; y = 1 + (r >> 1); x = (r & 1) * 34; }
  int idx = plane * PPL + y * 35 + x;
  G[idx] = 0.0f; H[idx] = 0.0f;
}
__global__ void k_init_gamma(float* __restrict__ gamma) {
  if (threadIdx.x == 0 && blockIdx.x == 0) gamma[0] = 1.0f;
}

// ================================================================ GEMMs
// PhiTPhi = Phip^T @ Phip.  Pad rows/cols of the result are exactly zero because
// the pad columns of Phip are zero -> store the full tile unconditionally.
__global__ void k_syrk(const float* __restrict__ Phip, float* __restrict__ out) {
  const int lane = threadIdx.x, lm = lane & 15, half = lane >> 4;
  const int t = blockIdx.x, ti = t / NT, tj = t % NT;
  const int ca = ti * 16 + lm, cb = tj * 16 + lm;
  v8f acc0 = {}, acc1 = {};
#pragma unroll 2
  for (int k0 = 0; k0 < Mdim; k0 += 8) {
    int ka = k0 + half * 2;
    v2f a0, b0, a1, b1;
    a0.x = Phip[(ka + 0) * LDPHI + ca];
    a0.y = Phip[(ka + 1) * LDPHI + ca];
    b0.x = Phip[(ka + 0) * LDPHI + cb];
    b0.y = Phip[(ka + 1) * LDPHI + cb];
    a1.x = Phip[(ka + 4) * LDPHI + ca];
    a1.y = Phip[(ka + 5) * LDPHI + ca];
    b1.x = Phip[(ka + 4) * LDPHI + cb];
    b1.y = Phip[(ka + 5) * LDPHI + cb];
    acc0 = wmma4(a0, b0, acc0);
    acc1 = wmma4(a1, b1, acc1);
  }
  acc0 += acc1;
#pragma unroll
  for (int r = 0; r < 8; ++r)
    out[(ti * 16 + half * 8 + r) * LDPTP + cb] = acc0[r];
}

// PhiTb = Phix[256,272] @ Phip[272,1104]  -> out [256, LDX] (pad cols get zeros)
__global__ void k_gemm_phitb(const float* __restrict__ Phix,
                             const float* __restrict__ Phip, float* __restrict__ out) {
  const int lane = threadIdx.x, lm = lane & 15, half = lane >> 4;
  const int t = blockIdx.x, ti = t / NT, tj = t % NT;
  const int arow = ti * 16 + lm, bcol = tj * 16 + lm;
  v8f acc0 = {}, acc1 = {};
#pragma unroll 2
  for (int k0 = 0; k0 < Mdim; k0 += 8) {
    int ka = k0 + half * 2;
    v2f a0 = *(const v2f*)(Phix + arow * Mdim + ka);
    v2f a1 = *(const v2f*)(Phix + arow * Mdim + ka + 4);
    v2f b0, b1;
    b0.x = Phip[(ka + 0) * LDPHI + bcol];
    b0.y = Phip[(ka + 1) * LDPHI + bcol];
    b1.x = Phip[(ka + 4) * LDPHI + bcol];
    b1.y = Phip[(ka + 5) * LDPHI + bcol];
    acc0 = wmma4(a0, b0, acc0);
    acc1 = wmma4(a1, b1, acc1);
  }
  acc0 += acc1;
#pragma unroll
  for (int r = 0; r < 8; ++r)
    out[(ti * 16 + half * 8 + r) * LDX + bcol] = acc0[r];
}

// x0 = Phix[256,272] @ Qinitp^T (Qinitp:[1104,272], pad rows zero) -> [256, LDX]
__global__ void k_gemm_x0(const float* __restrict__ Phix,
                          const float* __restrict__ Qp, float* __restrict__ out) {
  const int lane = threadIdx.x, lm = lane & 15, half = lane >> 4;
  const int t = blockIdx.x, ti = t / NT, tj = t % NT;
  const int arow = ti * 16 + lm, bcol = tj * 16 + lm;
  v8f acc0 = {}, acc1 = {};
#pragma unroll 2
  for (int k0 = 0; k0 < Mdim; k0 += 8) {
    int ka = k0 + half * 2;
    v2f a0 = *(const v2f*)(Phix + arow * Mdim + ka);
    v2f a1 = *(const v2f*)(Phix + arow * Mdim + ka + 4);
    v2f b0 = *(const v2f*)(Qp + bcol * Mdim + ka);
    v2f b1 = *(const v2f*)(Qp + bcol * Mdim + ka + 4);
    acc0 = wmma4(a0, b0, acc0);
    acc1 = wmma4(a1, b1, acc1);
  }
  acc0 += acc1;
#pragma unroll
  for (int r = 0; r < 8; ++r)
    out[(ti * 16 + half * 8 + r) * LDX + bcol] = acc0[r];
}

// fused X @ PhiTPhi:  MODE 0: out = X - a*acc + a*PhiTb   MODE 1: out = acc - PhiTb + GR
template <int MODE>
__global__ void k_xgemm(const float* __restrict__ X, const float* __restrict__ PTP,
                        const float* __restrict__ PhiTb, const float* __restrict__ GR,
                        const float* __restrict__ alphas, int p, float* __restrict__ out) {
  const int lane = threadIdx.x, lm = lane & 15, half = lane >> 4;
  const int t = blockIdx.x, ti = t / NT, tj = t % NT;
  const int arow = ti * 16 + lm, bcol = tj * 16 + lm;
  v8f acc0 = {}, acc1 = {};
#pragma unroll 2
  for (int k0 = 0; k0 < 1088; k0 += 8) {                  // 136 dual steps
    int ka = k0 + half * 2;
    v2f a0 = *(const v2f*)(X + arow * LDX + ka);
    v2f a1 = *(const v2f*)(X + arow * LDX + ka + 4);
    v2f b0, b1;
    b0.x = PTP[(ka + 0) * LDPTP + bcol];
    b0.y = PTP[(ka + 1) * LDPTP + bcol];
    b1.x = PTP[(ka + 4) * LDPTP + bcol];
    b1.y = PTP[(ka + 5) * LDPTP + bcol];
    acc0 = wmma4(a0, b0, acc0);
    acc1 = wmma4(a1, b1, acc1);
  }
  {                                                       // tail k = 1088..1091
    int ka = 1088 + half * 2;
    v2f a0 = *(const v2f*)(X + arow * LDX + ka);          // X pad cols are zero
    v2f b0;
    b0.x = PTP[(ka + 0) * LDPTP + bcol];                  // PTP pad rows are zero
    b0.y = PTP[(ka + 1) * LDPTP + bcol];
    acc0 = wmma4(a0, b0, acc0);
  }
  acc0 += acc1;
  float alpha = fabsf(alphas[p]);
#pragma unroll
  for (int r = 0; r < 8; ++r) {
    int idx = (ti * 16 + half * 8 + r) * LDX + bcol;
    if (MODE == 0)
      out[idx] = X[idx] - alpha * acc0[r] + alpha * PhiTb[idx];
    else
      out[idx] = acc0[r] - PhiTb[idx] + GR[idx];
  }
}

// ================================================================ conv (32ch->32ch)
// in/out/der: padded slabs [B*CH, 35, 35]; borders of `in` are zero (=zero padding).
// K order: k = tap*32 + ic ; tap is wave-uniform per k-step.
template <bool TRANSPOSED, bool WITH_DER>
__global__ void k_conv32(const float* __restrict__ in, const float* __restrict__ W,
                         const float* __restrict__ der, float* __restrict__ out) {
  const int lane = threadIdx.x & 31, lm = lane & 15, half = lane >> 4;
  const int oc0 = (threadIdx.x >> 5) * 16;                // 2 waves = 2 oc tiles
  const int wrow = oc0 + lm;
  int base[4], wb[4];
#pragma unroll
  for (int s = 0; s < 4; ++s) {
    int p = blockIdx.x * 64 + s * 16 + lm;                // NPIX % 64 == 0
    int b = p / Ndim, rem = p % Ndim;
    int y = rem / IMG, x = rem - y * IMG;
    base[s] = b * CH * PPL + y * 35 + x;
    wb[s]   = b * CH * PPL + (y + 1) * 35 + (x + 1);
  }
  v8f acc[4] = {};
#pragma unroll 2
  for (int k0 = 0; k0 < KCONV; k0 += 4) {
    const int tap = k0 >> 5;                              // wave-uniform
    const int ky = tap / 3, kx = tap - ky * 3;
    const int toff = ky * 35 + kx;
    const int ic0 = (k0 & 31) + half * 2;
    v2f av;
    if (!TRANSPOSED) {
      int wa = wrow * KCONV + ic0 * 9 + tap;
      av.x = W[wa]; av.y = W[wa + 9];
    } else {
      int wa = ic0 * KCONV + wrow * 9 + (8 - tap);
      av.x = W[wa]; av.y = W[wa + KCONV];
    }
    v2f bv[4];
#pragma unroll
    for (int s = 0; s < 4; ++s) {
      int ia = base[s] + ic0 * PPL + toff;
      bv[s].x = in[ia];
      bv[s].y = in[ia + PPL];
    }
#pragma unroll
    for (int s = 0; s < 4; ++s) acc[s] = wmma4(av, bv[s], acc[s]);
  }
#pragma unroll
  for (int r = 0; r < 8; ++r) {
    int oc = oc0 + half * 8 + r;
#pragma unroll
    for (int s = 0; s < 4; ++s) {
      int idx = wb[s] + oc * PPL;
      float v = acc[s][r];
      if (WITH_DER) v *= actderf(der[idx]);
      out[idx] = v;
    }
  }
}

// act() elementwise into padded slab (interior only; borders stay zero)
__global__ void k_act(const float* __restrict__ T, float* __restrict__ A) {
  int i = blockIdx.x * blockDim.x + threadIdx.x;          // B*CH*1089
  if (i >= B_SZ * CH * Ndim) return;
  int plane = i / Ndim, rem = i % Ndim;
  int y = rem / IMG, x = rem - y * IMG;
  int idx = plane * PPL + (y + 1) * 35 + (x + 1);
  A[idx] = actf(T[idx]);
}

// conv1 forward: 1ch [256,LDX] -> 32ch padded slab
__global__ void k_conv1_fwd(const float* __restrict__ Z, const float* __restrict__ c1,
                            float* __restrict__ out) {
  int p = blockIdx.x * blockDim.x + threadIdx.x;
  if (p >= NPIX) return;
  int b = p / Ndim, rem = p % Ndim, y = rem / IMG, x = rem - y * IMG;
  float inv[9];
#pragma unroll
  for (int t = 0; t < 9; ++t) {
    int ky = t / 3, kx = t - ky * 3;
    int yy = y + ky - 1, xx = x + kx - 1;
    bool ok = ((unsigned)yy < 33u) && ((unsigned)xx < 33u);
    int yc = min(max(yy, 0), 32), xc = min(max(xx, 0), 32);
    float v = Z[b * LDX + yc * IMG + xc];                 // unconditional (clamped)
    inv[t] = ok ? v : 0.0f;
  }
  int wbase = b * CH * PPL + (y + 1) * 35 + (x + 1);
  for (int oc = 0; oc < CH; ++oc) {
    float s = 0.0f;
#pragma unroll
    for (int t = 0; t < 9; ++t) s = fmaf(c1[oc * 9 + t], inv[t], s);
    out[wbase + oc * PPL] = s;
  }
}

// final convT with c1: padded slab 32ch -> [256,LDX] (borders of H are zero)
__global__ void k_convT1(const float* __restrict__ H, const float* __restrict__ c1,
                         float* __restrict__ out) {
  int p = blockIdx.x * blockDim.x + threadIdx.x;
  if (p >= NPIX) return;
  int b = p / Ndim, rem = p % Ndim, y = rem / IMG, x = rem - y * IMG;
  int base = b * CH * PPL + y * 35 + x;
  float s = 0.0f;
  for (int c = 0; c < CH; ++c) {
#pragma unroll
    for (int t = 0; t < 9; ++t) {
      int ky = t / 3, kx = t - ky * 3;
      s = fmaf(c1[c * 9 + (8 - t)], H[base + c * PPL + ky * 35 + kx], s);
    }
  }
  out[b * LDX + rem] = s;
}

// g_factor in place on padded G
__global__ void k_gfactor(float* __restrict__ g, const float* __restrict__ soft,
                          const float* __restrict__ gamma) {
  int p = blockIdx.x * blockDim.x + threadIdx.x;
  if (p >= NPIX) return;
  int b = p / Ndim, rem = p % Ndim, y = rem / IMG, x = rem - y * IMG;
  int base = b * CH * PPL + (y + 1) * 35 + (x + 1);
  float thr = soft[0] * gamma[0];
  float ss = 0.0f;
  for (int c = 0; c < CH; ++c) { float v = g[base + c * PPL]; ss += v * v; }
  float norm = sqrtf(ss);
  float scale = (norm > thr) ? 1.0f / fmaxf(norm, 1e-12f) : 1.0f / thr;
  for (int c = 0; c < CH; ++c) g[base + c * PPL] *= scale;
}

// x = z - tau * gr
__global__ void k_update_x(const float* __restrict__ Z, const float* __restrict__ GR,
                           const float* __restrict__ alphas, const float* __restrict__ betas,
                           int p, float* __restrict__ X) {
  int i = blockIdx.x * blockDim.x + threadIdx.x;
  if (i >= NPIX) return;
  int row = i / Ndim, col = i % Ndim, idx = row * LDX + col;
  float a = fabsf(alphas[p]), b = fabsf(betas[p]);
  float tau = a * b / (a + b);
  X[idx] = Z[idx] - tau * GR[idx];
}

__global__ void k_rownorm(const float* __restrict__ G, float* __restrict__ norms) {
  __shared__ float sm[256];
  int b = blockIdx.x, tid = threadIdx.x;
  float s = 0.0f;
  for (int j = tid; j < Ndim; j += 256) { float v = G[b * LDX + j]; s += v * v; }
  sm[tid] = s; __syncthreads();
  for (int off = 128; off > 0; off >>= 1) {
    if (tid < off) sm[tid] += sm[tid + off];
    __syncthreads();
  }
  if (tid == 0) norms[b] = sqrtf(sm[0]);
}

__global__ void k_gamma(const float* __restrict__ norms, const float* __restrict__ soft,
                        float* __restrict__ gamma) {
  __shared__ float sm[256];
  int tid = threadIdx.x;
  sm[tid] = norms[tid]; __syncthreads();
  for (int off = 128; off > 0; off >>= 1) {
    if (tid < off) sm[tid] += sm[tid + off];
    __syncthreads();
  }
  if (tid == 0) {
    float mean = sm[0] * (1.0f / 256.0f);
    float g = gamma[0];
    gamma[0] = g * ((mean < SIGMA_C * g * soft[0]) ? 0.9f : 1.0f);
  }
}

__global__ void k_copy_out(const float* __restrict__ X, float* __restrict__ dst) {
  int i = blockIdx.x * blockDim.x + threadIdx.x;
  if (i >= NPIX) return;
  int row = i / Ndim, col = i % Ndim;
  dst[i] = X[row * LDX + col];
}

// ================================================================ host side
static void grad_r_seq(const float* src, float* GR, float* T1, float* T2, float* T3,
                       float* G, float* H, const float* c1, const float* c2,
                       const float* c3, const float* c4, const float* soft,
                       const float* gamma, hipStream_t s) {
  dim3 cg(NPIX / 64), cb(64);
  const int eb = 256;
  const int eg  = (NPIX + eb - 1) / eb;
  const int ag  = (B_SZ * CH * Ndim + eb - 1) / eb;
  k_conv1_fwd<<<eg, eb, 0, s>>>(src, c1, T1);
  k_act<<<ag, eb, 0, s>>>(T1, H);
  k_conv32<false, false><<<cg, cb, 0, s>>>(H, c2, nullptr, T2);
  k_act<<<ag, eb, 0, s>>>(T2, H);
  k_conv32<false, false><<<cg, cb, 0, s>>>(H, c3, nullptr, T3);
  k_act<<<ag, eb, 0, s>>>(T3, H);
  k_conv32<false, false><<<cg, cb, 0, s>>>(H, c4, nullptr, G);
  k_gfactor<<<eg, eb, 0, s>>>(G, soft, gamma);
  k_conv32<true, true><<<cg, cb, 0, s>>>(G, c4, T3, H);
  k_conv32<true, true><<<cg, cb, 0, s>>>(H, c3, T2, G);
  k_conv32<true, true><<<cg, cb, 0, s>>>(G, c2, T1, H);
  k_convT1<<<eg, eb, 0, s>>>(H, c1, GR);
}

extern "C" void kernel_launch(void* const* d_in, const int* in_sizes, int n_in,
                              void* d_out, int out_size, void* d_ws, size_t ws_size,
                              hipStream_t stream) {
  (void)in_sizes; (void)n_in; (void)out_size; (void)ws_size;
  const float* Phix   = (const float*)d_in[0];
  const float* Phi    = (const float*)d_in[1];
  const float* Qinit  = (const float*)d_in[2];
  const float* soft   = (const float*)d_in[3];
  const float* alphas = (const float*)d_in[4];
  const float* betas  = (const float*)d_in[5];
  const float* c1     = (const float*)d_in[6];
  const float* c2     = (const float*)d_in[7];
  const float* c3     = (const float*)d_in[8];
  const float* c4     = (const float*)d_in[9];

  float* ws = (float*)d_ws;
  size_t off = 0;
  const size_t PSLAB = (size_t)B_SZ * CH * PPL;           // 10,035,200 floats
  float* PTP   = ws + off; off += (size_t)RPTP * LDPTP;   // 1104*1120
  float* PHIP  = ws + off; off += (size_t)Mdim * LDPHI;
  float* QP    = ws + off; off += (size_t)LDPHI * Mdim;
  float* PHITB = ws + off; off += (size_t)B_SZ * LDX;
  float* X     = ws + off; off += (size_t)B_SZ * LDX;
  float* Z     = ws + off; off += (size_t)B_SZ * LDX;     // reused as gphi
  float* GR    = ws + off; off += (size_t)B_SZ * LDX;
  float* T1    = ws + off; off += PSLAB;
  float* T2    = ws + off; off += PSLAB;
  float* T3    = ws + off; off += PSLAB;
  float* G     = ws + off; off += PSLAB;
  float* H     = ws + off; off += PSLAB;
  float* NORMS = ws + off; off += 256;
  float* GAMMA = ws + off; off += 1;

  const int eb = 256;
  const int eg = (NPIX + eb - 1) / eb;

  // per-launch init (deterministic; ws is not re-poisoned between replays)
  k_init_gamma<<<1, 32, 0, stream>>>(GAMMA);
  k_pad_phi<<<(Mdim * LDPHI + eb - 1) / eb, eb, 0, stream>>>(Phi, PHIP);
  k_pad_qinit<<<(LDPHI * Mdim + eb - 1) / eb, eb, 0, stream>>>(Qinit, QP);
  k_zero_borders<<<(B_SZ * CH * 136 + eb - 1) / eb, eb, 0, stream>>>(G, H);

  k_syrk<<<NT * NT, 32, 0, stream>>>(PHIP, PTP);              // also zeroes PTP pads
  k_gemm_phitb<<<(B_SZ / 16) * NT, 32, 0, stream>>>(Phix, PHIP, PHITB);
  k_gemm_x0<<<(B_SZ / 16) * NT, 32, 0, stream>>>(Phix, QP, X); // also zeroes X pads

  for (int p = 0; p < 3; ++p) {
    k_xgemm<0><<<(B_SZ / 16) * NT, 32, 0, stream>>>(X, PTP, PHITB, nullptr, alphas, p, Z);
    grad_r_seq(Z, GR, T1, T2, T3, G, H, c1, c2, c3, c4, soft, GAMMA, stream);
    k_update_x<<<eg, eb, 0, stream>>>(Z, GR, alphas, betas, p, X);
    grad_r_seq(X, GR, T1, T2, T3, G, H, c1, c2, c3, c4, soft, GAMMA, stream);
    k_xgemm<1><<<(B_SZ / 16) * NT, 32, 0, stream>>>(X, PTP, PHITB, GR, alphas, p, Z);
    k_rownorm<<<B_SZ, 256, 0, stream>>>(Z, NORMS);
    k_gamma<<<1, 256, 0, stream>>>(NORMS, soft, GAMMA);
  }
  k_copy_out<<<eg, eb, 0, stream>>>(X, (float*)d_out);
}